// GraphAttentionLayer_18820546691890
// MI455X (gfx1250) — compile-verified
//
#include <hip/hip_runtime.h>

typedef __attribute__((ext_vector_type(2))) float v2f;
typedef __attribute__((ext_vector_type(4))) float v4f;
typedef __attribute__((ext_vector_type(8))) float v8f;

#if __has_builtin(__builtin_amdgcn_wmma_f32_16x16x4_f32)
#define GAT_USE_WMMA 1
#else
#define GAT_USE_WMMA 0
#endif

#define NEG_SLOPE 0.01f

// Broadcast lane srcLane's value to all lanes via v_readlane (SGPR), no LDS pipe.
__device__ __forceinline__ float readlane_f32(float v, int srcLane) {
    return __int_as_float(__builtin_amdgcn_readlane(__float_as_int(v), srcLane));
}

// One wave32 processes 4 nodes (= 16 rows of the flattened (4N x 256) score GEMV).
// Stage 1: scores via chained V_WMMA_F32_16X16X4_F32; the att chunk is replicated
//          across all 16 B columns so every lane holds the row dot products
//          (matrix pipe does the K-reduction; no shuffle tree).
//          A loads use default (RT) caching: each 128B line is consumed as 16B x 8
//          consecutive iterations, so L0 residency matters (NT would amplify L2 traffic).
// Stage 2: softmax(LeakyReLU(scores)) over the 4 neighbors, in registers (readlane bcast).
// Stage 3: weighted aggregation of c/a/p/v rows, float4 per lane, NT loads/stores
//          (each line touched exactly once -> stream past the caches).
__global__ __launch_bounds__(256) void gat_fused_kernel(
    const int* __restrict__ node_type,
    const float* __restrict__ c_agg,   // [N,128]
    const float* __restrict__ ce,      // [N,4,256] == flattened [4N,256]
    const float* __restrict__ a_agg,   // [N,128]
    const float* __restrict__ p_agg,   // [N,128]
    const float* __restrict__ v_agg,   // [N,128]
    const float* __restrict__ a_att,   // [256]
    const float* __restrict__ p_att,   // [256]
    const float* __restrict__ v_att,   // [256]
    float* __restrict__ out,           // [N,128]
    int N)
{
    const int lane  = threadIdx.x & 31;
    const int wave  = threadIdx.x >> 5;
    const int group = blockIdx.x * 8 + wave;     // 4 nodes per wave
    const int numGroups = (N + 3) >> 2;
    if (group >= numGroups) return;              // wave-uniform: EXEC stays all-1s for WMMA

    const int nt = node_type[0];
    const float* __restrict__ att = (nt == 1) ? a_att : ((nt == 2) ? p_att : v_att);

    // A-matrix fragment addressing (16x4 f32 A layout, ISA 7.12.2):
    //   lanes 0-15 : row M = lane,    K = 0,1 of each 4-chunk
    //   lanes 16-31: row M = lane-16, K = 2,3 of each 4-chunk
    const long rowBase = (long)group * 16;
    const long maxRow  = (long)N * 4 - 1;
    const int  half    = lane >> 4;
    long r = rowBase + (lane & 15);
    if (r > maxRow) r = maxRow;                  // tail clamp (unused stores are skipped)
    const float* __restrict__ aP = ce  + r * 256 + half * 2;
    const float* __restrict__ bP = att + half * 2;   // replicate chunk over all 16 B cols

#if GAT_USE_WMMA
    v8f acc = {0.0f, 0.0f, 0.0f, 0.0f, 0.0f, 0.0f, 0.0f, 0.0f};
#pragma unroll 8
    for (int kc = 0; kc < 64; ++kc) {
        v2f a = *(const v2f*)(aP + kc * 4);       // RT: line reused across 8 iterations
        v2f b = *(const v2f*)(bP + kc * 4);       // L0-resident, 1 KB total
        acc = __builtin_amdgcn_wmma_f32_16x16x4_f32(
            /*neg_a=*/false, a, /*neg_b=*/false, b,
            /*c_mod=*/(short)0, acc, /*reuse_a=*/false, /*reuse_b=*/false);
    }
    // D layout: lanes 0-15 hold rows rowBase+0..7 in acc[0..7]; lanes 16-31 rows +8..15.
#else
    // Fallback: same loads, lane-pair partial dot + one xor-shuffle to combine halves.
    float part = 0.0f;
#pragma unroll 8
    for (int kc = 0; kc < 64; ++kc) {
        v2f a = *(const v2f*)(aP + kc * 4);
        v2f b = *(const v2f*)(bP + kc * 4);
        part = fmaf(a.x, b.x, fmaf(a.y, b.y, part));
    }
    const float dotr = part + __shfl_xor(part, 16, 32);  // lane l holds row rowBase+(l&15)
#endif

    // Redistribute scores + softmax(LeakyReLU(.)) per node j (uniform across lanes).
    float w[4][4];
#pragma unroll
    for (int j = 0; j < 4; ++j) {
        float s[4];
#pragma unroll
        for (int k = 0; k < 4; ++k) {
#if GAT_USE_WMMA
            // row 4j+k: element 4*(j&1)+k, from lane 0 (rows<8) or lane 16 (rows>=8)
            s[k] = readlane_f32(acc[4 * (j & 1) + k], (j & 2) ? 16 : 0);
#else
            s[k] = readlane_f32(dotr, 4 * j + k);
#endif
            s[k] = (s[k] >= 0.0f) ? s[k] : NEG_SLOPE * s[k];
        }
        const float m  = fmaxf(fmaxf(s[0], s[1]), fmaxf(s[2], s[3]));
        const float e0 = __expf(s[0] - m);
        const float e1 = __expf(s[1] - m);
        const float e2 = __expf(s[2] - m);
        const float e3 = __expf(s[3] - m);
        const float inv = 1.0f / (e0 + e1 + e2 + e3);
        w[j][0] = e0 * inv; w[j][1] = e1 * inv; w[j][2] = e2 * inv; w[j][3] = e3 * inv;
    }

    // Weighted aggregation: lane owns floats [4*lane, 4*lane+4) of each 128-wide row.
    const int d0 = lane << 2;
#pragma unroll
    for (int j = 0; j < 4; ++j) {
        const long n = (long)group * 4 + j;
        if (n >= N) break;                       // wave-uniform
        const long off = n * 128 + d0;
        v4f cv = __builtin_nontemporal_load((const v4f*)(c_agg + off));
        v4f av = __builtin_nontemporal_load((const v4f*)(a_agg + off));
        v4f pv = __builtin_nontemporal_load((const v4f*)(p_agg + off));
        v4f vv = __builtin_nontemporal_load((const v4f*)(v_agg + off));
        v4f o = w[j][0] * cv + w[j][1] * av + w[j][2] * pv + w[j][3] * vv;
        __builtin_nontemporal_store(o, (v4f*)(out + off));
    }
}

extern "C" void kernel_launch(void* const* d_in, const int* in_sizes, int n_in,
                              void* d_out, int out_size, void* d_ws, size_t ws_size,
                              hipStream_t stream) {
    const int* node_type = (const int*)d_in[0];
    const float* c_agg   = (const float*)d_in[1];
    const float* ce      = (const float*)d_in[2];
    const float* a_agg   = (const float*)d_in[3];
    const float* p_agg   = (const float*)d_in[4];
    const float* v_agg   = (const float*)d_in[5];
    const float* a_att   = (const float*)d_in[6];
    const float* p_att   = (const float*)d_in[7];
    const float* v_att   = (const float*)d_in[8];

    const int N = in_sizes[1] / 128;             // c_agg_batch is [N,128]
    const int numGroups = (N + 3) / 4;           // 4 nodes per wave
    const int blocks = (numGroups + 7) / 8;      // 8 waves (256 threads) per block

    gat_fused_kernel<<<blocks, 256, 0, stream>>>(
        node_type, c_agg, ce, a_agg, p_agg, v_agg,
        a_att, p_att, v_att, (float*)d_out, N);
}